// similarity_for_visualize_71193377898743
// MI455X (gfx1250) — compile-verified
//
#include <hip/hip_runtime.h>
#include <hip/hip_bf16.h>
#include <math.h>

typedef __attribute__((ext_vector_type(2))) float v2f;
typedef __attribute__((ext_vector_type(8))) float v8f;

// ---------------- problem constants (specialized to the reference inputs) ---
#define NB   8      // batch (M == N == 8)
#define CCH  32     // channels
#define GH   32     // grd spatial
#define GW   32
#define GK   32768  // C*GH*GW  reduction length
#define WS   128    // sat full width/height
#define CROP 16     // sat crop offset
#define HI   96     // cropped sat spatial
#define HO   65     // output spatial (96-32+1)
#define NPOS 4225   // 65*65
#define S2N  9216   // 96*96

// GEMM tiling
#define TPW     4                   // 16-wide position tiles per wave
#define KSLICES 4                   // K split across grid.z
#define CPS     (CCH / KSLICES)     // channels per slice = 8
#define NTG     ((NPOS + 63) / 64)  // tile groups = 67
#define CORRSZ  (NB * NB * NPOS)    // 270400

// ---------------- workspace layout (floats) --------------------------------
#define WS_S2      0
#define WS_ROWSUM  (WS_S2      + NB * S2N)          // 73728
#define WS_PART    (WS_ROWSUM  + NB * HI * HO)      // +49920
#define WS_CORR    (WS_PART    + NB * NPOS)         // +33800
#define WS_CORRP   (WS_CORR    + CORRSZ)            // +270400
#define WS_GNORM   (WS_CORRP   + KSLICES * CORRSZ)  // +1081600
#define WS_SNORM   (WS_GNORM   + NB)
#define WS_GCS     (WS_SNORM   + NB)                // grd channel sums, 8192

// ---------------------------------------------------------------------------
__global__ void zero_kernel(float* p, int n) {
  int i = blockIdx.x * blockDim.x + threadIdx.x;
  if (i < n) p[i] = 0.0f;
}

// per-filter L2 norm of grd (8 blocks x 256)
__global__ __launch_bounds__(256) void gnorm_kernel(const float* __restrict__ grd,
                                                    float* __restrict__ gnorm) {
  int n = blockIdx.x;
  const float* g = grd + n * GK;
  float s = 0.f;
  for (int i = threadIdx.x; i < GK; i += 256) { float v = g[i]; s += v * v; }
  __shared__ float sm[256];
  sm[threadIdx.x] = s; __syncthreads();
  for (int off = 128; off > 0; off >>= 1) {
    if (threadIdx.x < off) sm[threadIdx.x] += sm[threadIdx.x + off];
    __syncthreads();
  }
  if (threadIdx.x == 0) gnorm[n] = fmaxf(sqrtf(sm[0]), 1e-12f);
}

// s2[m][y][x] = sum_c sat_crop^2, plus per-image norm (8 blocks x 256)
__global__ __launch_bounds__(256) void s2_kernel(const float* __restrict__ sat,
                                                 float* __restrict__ s2,
                                                 float* __restrict__ snorm) {
  int m = blockIdx.x;
  const float* base = sat + (size_t)m * CCH * WS * WS;
  float tot = 0.f;
  for (int p = threadIdx.x; p < S2N; p += 256) {
    int y = p / HI, x = p % HI;
    const float* pix = base + (CROP + y) * WS + (CROP + x);
    float s = 0.f;
    #pragma unroll
    for (int c = 0; c < CCH; ++c) { float v = pix[c * WS * WS]; s += v * v; }
    s2[m * S2N + p] = s;
    tot += s;
  }
  __shared__ float sm[256];
  sm[threadIdx.x] = tot; __syncthreads();
  for (int off = 128; off > 0; off >>= 1) {
    if (threadIdx.x < off) sm[threadIdx.x] += sm[threadIdx.x + off];
    __syncthreads();
  }
  if (threadIdx.x == 0) snorm[m] = fmaxf(sqrtf(sm[0]), 1e-12f);
}

// channel sums of raw grd (for the mask)
__global__ void gchansum_kernel(const float* __restrict__ grd, float* __restrict__ gcs) {
  int idx = blockIdx.x * blockDim.x + threadIdx.x;
  if (idx >= NB * GH * GW) return;
  int n = idx / (GH * GW), hw = idx % (GH * GW);
  const float* g = grd + n * GK + hw;
  float s = 0.f;
  #pragma unroll
  for (int c = 0; c < CCH; ++c) s += g[c * GH * GW];
  gcs[idx] = s;
}

// separable 32-wide window sum along x
__global__ void rowsum_kernel(const float* __restrict__ s2, float* __restrict__ rs) {
  int idx = blockIdx.x * blockDim.x + threadIdx.x;
  if (idx >= NB * HI * HO) return;
  int m = idx / (HI * HO), r = idx % (HI * HO);
  int y = r / HO, ow = r % HO;
  const float* row = s2 + m * S2N + y * HI + ow;
  float s = 0.f;
  #pragma unroll
  for (int j = 0; j < GW; ++j) s += row[j];
  rs[idx] = s;
}

// then 32-tall window sum along y -> partical_raw
__global__ void partical_kernel(const float* __restrict__ rs, float* __restrict__ part) {
  int idx = blockIdx.x * blockDim.x + threadIdx.x;
  if (idx >= NB * NPOS) return;
  int m = idx / NPOS, r = idx % NPOS;
  int oh = r / HO, ow = r % HO;
  const float* col = rs + m * HI * HO + oh * HO + ow;
  float s = 0.f;
  #pragma unroll
  for (int i = 0; i < GH; ++i) s += col[i * HO];
  part[idx] = s;
}

// -------------------- WMMA f32 cross-correlation ---------------------------
// grid (67 tile-groups, 8 images, 4 K-slices); one wave per block.
// Each wave: 4 position tiles share one A (grd) operand per K-step ->
// 1 A load + 4 B loads per 4 WMMAs; 4 independent accumulation chains.
__global__ __launch_bounds__(32) void corr_wmma_kernel(const float* __restrict__ grd,
                                                       const float* __restrict__ sat,
                                                       float* __restrict__ corrp) {
  const int tg   = blockIdx.x;           // 0..66 (64 positions each)
  const int m    = blockIdx.y;           // 0..7
  const int ks   = blockIdx.z;           // 0..3 (8 channels each)
  const int lane = threadIdx.x;
  const int half = lane >> 4;            // 0 | 1
  const int lid  = lane & 15;

  const float* satm = sat + (size_t)m * CCH * WS * WS
                          + (size_t)ks * CPS * WS * WS;   // this slice's channels

  int  p[TPW];
  bool pv[TPW];
  const float* bBase[TPW];
  #pragma unroll
  for (int t = 0; t < TPW; ++t) {
    int q = tg * 64 + t * 16 + lid;
    pv[t] = (q < NPOS);
    if (q >= NPOS) q = NPOS - 1;         // clamp (duplicate work, not stored)
    p[t] = q;
    int oh = q / HO, ow = q % HO;
    bBase[t] = satm + (CROP + oh) * WS + (CROP + ow);
  }

  // A operand: row n = lid (rows 8..15 alias filter 0; results discarded)
  const int n = (lid < 8) ? lid : 0;
  const float* aPtr = grd + n * GK + ks * CPS * GH * GW + half * 2;  // 8B aligned

  v8f acc[TPW];
  #pragma unroll
  for (int t = 0; t < TPW; ++t) acc[t] = (v8f){0.f,0.f,0.f,0.f,0.f,0.f,0.f,0.f};

  for (int c = 0; c < CPS; ++c) {
    const float* aC = aPtr + c * GH * GW;
    for (int i = 0; i < GH; ++i) {
      const float* aRow = aC + i * GW;
      const int rowOff = c * WS * WS + i * WS;
      #pragma unroll
      for (int j = 0; j < GW; j += 4) {
        v2f a = *(const v2f*)(aRow + j);          // {K,K+1} ({K+2,K+3} upper half)
        #pragma unroll
        for (int t = 0; t < TPW; ++t) {
          const float* bRow = bBase[t] + rowOff;
          v2f b; b.x = bRow[j + half * 2];
                 b.y = bRow[j + half * 2 + 1];
          acc[t] = __builtin_amdgcn_wmma_f32_16x16x4_f32(false, a, false, b,
                                                         (short)0, acc[t],
                                                         false, false);
        }
      }
    }
  }

  // C/D layout: VGPR r, lanes 0-15 -> row M=r (filter n=r); upper half rows
  // 8..15 are padding filters -> only lower half stores.
  if (half == 0) {
    float* dstm = corrp + ((size_t)ks * NB + m) * NB * NPOS;
    #pragma unroll
    for (int t = 0; t < TPW; ++t) {
      if (pv[t]) {
        float* dst = dstm + p[t];
        #pragma unroll
        for (int r = 0; r < 8; ++r) dst[r * NPOS] = acc[t][r];
      }
    }
  }
}

// sum the 4 K-slice partials
__global__ void corr_reduce_kernel(const float* __restrict__ corrp,
                                   float* __restrict__ corr) {
  int idx = blockIdx.x * blockDim.x + threadIdx.x;
  if (idx >= CORRSZ) return;
  corr[idx] = (corrp[idx] + corrp[idx + CORRSZ])
            + (corrp[idx + 2 * CORRSZ] + corrp[idx + 3 * CORRSZ]);
}

// -------------------- finalize: similarity = max over positions ------------
__global__ __launch_bounds__(256) void sim_kernel(const float* __restrict__ corr,
                                                  const float* __restrict__ part,
                                                  const float* __restrict__ gnorm,
                                                  const float* __restrict__ snorm,
                                                  float* __restrict__ out) {
  int m = blockIdx.x / NB, n = blockIdx.x % NB;
  const float* c  = corr + (size_t)(m * NB + n) * NPOS;
  const float* pt = part + (size_t)m * NPOS;
  float sn = snorm[m], gn = gnorm[n];
  float inv = 1.f / (sn * gn);
  float inv_sn2 = 1.f / (sn * sn);
  float best = -INFINITY;
  for (int p = threadIdx.x; p < NPOS; p += 256) {
    float denom = fmaxf(pt[p] * inv_sn2, 1e-12f);
    best = fmaxf(best, c[p] * inv / denom);
  }
  __shared__ float sm[256];
  sm[threadIdx.x] = best; __syncthreads();
  for (int off = 128; off > 0; off >>= 1) {
    if (threadIdx.x < off) sm[threadIdx.x] = fmaxf(sm[threadIdx.x], sm[threadIdx.x + off]);
    __syncthreads();
  }
  if (threadIdx.x == 0) out[m * NB + n] = sm[0];
}

// ---- diag argmax (first occurrence) + mask placement, one block per m -----
__global__ __launch_bounds__(256) void diag_mask_kernel(const float* __restrict__ corr,
                                                        const float* __restrict__ part,
                                                        const float* __restrict__ gnorm,
                                                        const float* __restrict__ snorm,
                                                        const float* __restrict__ gcs,
                                                        float* __restrict__ out_mask) {
  int m = blockIdx.x;
  int tid = threadIdx.x;
  const float* c  = corr + (size_t)(m * NB + m) * NPOS;
  const float* pt = part + (size_t)m * NPOS;
  float sn = snorm[m], gn = gnorm[m];
  float inv = 1.f / (sn * gn);
  float inv_sn2 = 1.f / (sn * sn);

  float best = -INFINITY; int bi = 0x7fffffff;
  for (int p = tid; p < NPOS; p += 256) {
    float denom = fmaxf(pt[p] * inv_sn2, 1e-12f);
    float v = c[p] * inv / denom;
    if (v > best) { best = v; bi = p; }          // strided scan keeps lowest p per thread
  }
  __shared__ float sv[256]; __shared__ int si[256];
  sv[tid] = best; si[tid] = bi; __syncthreads();
  for (int off = 128; off > 0; off >>= 1) {
    if (tid < off) {
      float vo = sv[tid + off]; int io = si[tid + off];
      if (vo > sv[tid] || (vo == sv[tid] && io < si[tid])) { sv[tid] = vo; si[tid] = io; }
    }
    __syncthreads();
  }
  int pos = si[0];
  int ph = pos / HO, pw = pos % HO;
  // grd_mask = |sum_c grd_normalized| > 1e-6  ==  |gcs|/gnorm > 1e-6
  for (int q = tid; q < GH * GW; q += 256) {
    int h = q / GW, w = q % GW;
    float mv = (fabsf(gcs[m * GH * GW + q]) / gn > 1e-6f) ? 1.f : 0.f;
    out_mask[(size_t)m * S2N + (ph + h) * HI + (pw + w)] = mv;
  }
}

// ---------------------------------------------------------------------------
extern "C" void kernel_launch(void* const* d_in, const int* in_sizes, int n_in,
                              void* d_out, int out_size, void* d_ws, size_t ws_size,
                              hipStream_t stream) {
  const float* grd = (const float*)d_in[0];   // (8,32,32,32)
  const float* sat = (const float*)d_in[1];   // (8,32,128,128)
  float* out = (float*)d_out;                 // 64 sim + 73728 mask
  float* ws  = (float*)d_ws;

  float* s2     = ws + WS_S2;
  float* rowsum = ws + WS_ROWSUM;
  float* part   = ws + WS_PART;
  float* corr   = ws + WS_CORR;
  float* corrp  = ws + WS_CORRP;
  float* gnorm  = ws + WS_GNORM;
  float* snorm  = ws + WS_SNORM;
  float* gcs    = ws + WS_GCS;

  const int total_out = NB * NB + NB * S2N;   // 73792
  zero_kernel<<<(total_out + 255) / 256, 256, 0, stream>>>(out, total_out);

  gnorm_kernel<<<NB, 256, 0, stream>>>(grd, gnorm);
  s2_kernel<<<NB, 256, 0, stream>>>(sat, s2, snorm);
  gchansum_kernel<<<(NB * GH * GW + 255) / 256, 256, 0, stream>>>(grd, gcs);

  rowsum_kernel<<<(NB * HI * HO + 255) / 256, 256, 0, stream>>>(s2, rowsum);
  partical_kernel<<<(NB * NPOS + 255) / 256, 256, 0, stream>>>(rowsum, part);

  dim3 cgrid(NTG, NB, KSLICES);               // 67 x 8 x 4 waves
  corr_wmma_kernel<<<cgrid, 32, 0, stream>>>(grd, sat, corrp);
  corr_reduce_kernel<<<(CORRSZ + 255) / 256, 256, 0, stream>>>(corrp, corr);

  sim_kernel<<<NB * NB, 256, 0, stream>>>(corr, part, gnorm, snorm, out);
  diag_mask_kernel<<<NB, 256, 0, stream>>>(corr, part, gnorm, snorm, gcs,
                                           out + NB * NB);
}